// CapsuleLayer_36472862277927
// MI455X (gfx1250) — compile-verified
//
#include <hip/hip_runtime.h>
#include <math.h>

typedef __attribute__((ext_vector_type(2))) float v2f;
typedef __attribute__((ext_vector_type(8))) float v8f;

#define R_NODES 1152
#define CI 8
#define CO 16
#define NCAPS 10
#define BATCH 256
#define BT 16                      // batch tile = WMMA M
#define WAVES 8
#define RCHUNK (R_NODES / WAVES)   // 144 route nodes per wave
#define SEGS 16
#define RSEG (R_NODES / SEGS)      // 72

// D = A(16x4 f32) * B(4x16 f32) + C(16x16 f32); exact f32 WMMA, chained accumulator.
static __device__ __forceinline__ v8f wmma_f32_16x16x4(v2f a, v2f b, v8f c) {
  return __builtin_amdgcn_wmma_f32_16x16x4_f32(false, a, false, b, (short)0, c,
                                               false, false);
}

// v += row_ror:<N>(v)  -- DPP16 row rotate + add, stays on the VALU pipe
// (no LDS/ds_swizzle traffic). Row = 16 lanes = one half of the wave32,
// which is exactly the 16-output-column reduction domain of the C fragment.
template <int CTRL>
static __device__ __forceinline__ float dpp_ror_add(float v) {
  const int p =
      __builtin_amdgcn_update_dpp(0, __float_as_int(v), CTRL, 0xf, 0xf, false);
  return v + __int_as_float(p);
}
static __device__ __forceinline__ float row_sum16(float v) {
  v = dpp_ror_add<0x128>(v);   // row_ror:8
  v = dpp_ror_add<0x124>(v);   // row_ror:4
  v = dpp_ror_add<0x122>(v);   // row_ror:2
  v = dpp_ror_add<0x121>(v);   // row_ror:1  -> every lane holds the row total
  return v;
}

__global__ __launch_bounds__(256)
void capsule_route_kernel(const float* __restrict__ x,
                          const float* __restrict__ rw,
                          float* __restrict__ out) {
  // All routing state lives in LDS (~84 KB of 320 KB/WGP); priors are never
  // materialized -- x/W (15 MB total) stay L2-resident across the 5 passes.
  __shared__ float logits[BT * R_NODES];   // 73728 B : per-(b,r) routing logits
  __shared__ float sred[WAVES * 256];      //  8192 B : cross-wave accumulator reduce
  __shared__ float sout[BT * CO];          //  1024 B : squashed output tile
  __shared__ float stmp[SEGS * BT];        //  1024 B : softmax partials
  __shared__ float smax[BT];
  __shared__ float sinv[BT];

  const int tid  = threadIdx.x;
  const int lane = tid & 31;
  const int wave = tid >> 5;
  const int col  = lane & 15;   // A: batch row m; B: output col o; C: N col
  const int grp  = lane >> 4;   // K half (K 0..1 vs 2..3 per WMMA layout)
  const int klo  = grp * 2;

  const int n  = blockIdx.x / (BATCH / BT);
  const int bt = blockIdx.x % (BATCH / BT);
  const int b0 = bt * BT;

  const int r0 = wave * RCHUNK;
  const int r1 = r0 + RCHUNK;

  const float* xbase = x + (size_t)(b0 + col) * R_NODES * CI;      // x[b, r, c]
  const float* wbase = rw + (size_t)n * R_NODES * CI * CO + col;   // W[n, r, c, o=col]

  // ---- fragment loads for one route node r ------------------------------
  auto load_frags = [&](int r, v2f& a0, v2f& a1, v2f& bw0, v2f& bw1) {
    const float2* xv = (const float2*)(xbase + (size_t)r * CI);
    float2 lo = xv[grp];        // x[b, r, klo..klo+1]
    float2 hi = xv[2 + grp];    // x[b, r, klo+4..klo+5]
    a0.x = lo.x; a0.y = lo.y;
    a1.x = hi.x; a1.y = hi.y;
    const float* wp = wbase + (size_t)r * (CI * CO);
    bw0.x = wp[klo * CO];       bw0.y = wp[(klo + 1) * CO];
    bw1.x = wp[(klo + 4) * CO]; bw1.y = wp[(klo + 5) * CO];
  };

  // ---- weighted-sum pass: s[b,o] = sum_r p[b,r]*priors[b,r,o]  ----------
  // One long GEMM over K = R*Ci with p folded into the A fragment.
  // `uniform` is a literal at each call site -> branch-free specialized loops.
  auto sum_pass = [&](bool uniform) -> v8f {
    float pmax = 0.f, pinv = 0.f;
    if (!uniform) { pmax = smax[col]; pinv = sinv[col]; }
    v8f acc = {0.f, 0.f, 0.f, 0.f, 0.f, 0.f, 0.f, 0.f};
#pragma unroll 4
    for (int r = r0; r < r1; ++r) {
      v2f a0, a1, bw0, bw1;
      load_frags(r, a0, a1, bw0, bw1);
      const float p = uniform
          ? (1.0f / (float)R_NODES)
          : __expf(logits[col * R_NODES + r] - pmax) * pinv;
      a0.x *= p; a0.y *= p; a1.x *= p; a1.y *= p;
      acc = wmma_f32_16x16x4(a0, bw0, acc);   // K = 0..3 of this route node
      acc = wmma_f32_16x16x4(a1, bw1, acc);   // K = 4..7
    }
    return acc;
  };

  // ---- cross-wave reduce + squash into sout -----------------------------
  auto reduce_squash = [&](v8f acc) {
#pragma unroll
    for (int j = 0; j < 8; ++j) sred[wave * 256 + j * 32 + lane] = acc[j];
    __syncthreads();
    {
      float v = 0.f;
#pragma unroll
      for (int w = 0; w < WAVES; ++w) v += sred[w * 256 + tid];
      const int jj = tid >> 5, ll = tid & 31;
      sout[(jj + 8 * (ll >> 4)) * CO + (ll & 15)] = v;   // C frag -> [b][o]
    }
    __syncthreads();
    if (tid < BT) {   // squash: v *= |v|^2 / ((1+|v|^2)*|v|)
      float sq = 0.f;
#pragma unroll
      for (int o = 0; o < CO; ++o) { float u = sout[tid * CO + o]; sq += u * u; }
      const float scale = sq / ((1.0f + sq) * sqrtf(sq));
#pragma unroll
      for (int o = 0; o < CO; ++o) sout[tid * CO + o] *= scale;
    }
    __syncthreads();
  };

  // ---- delta pass + softmax stats ---------------------------------------
  // logits[b][r] (+)= dot_o(priors[b,r,:], out[b,:]); running max fused in.
  auto delta_pass = [&](bool first) {
    float outv[8];
#pragma unroll
    for (int j = 0; j < 8; ++j) outv[j] = sout[(j + 8 * grp) * CO + col];
    float lmax[8];
#pragma unroll
    for (int j = 0; j < 8; ++j) lmax[j] = -3.4e38f;
    const v8f vzero = {0.f, 0.f, 0.f, 0.f, 0.f, 0.f, 0.f, 0.f};
#pragma unroll 2
    for (int r = r0; r < r1; ++r) {
      v2f a0, a1, bw0, bw1;
      load_frags(r, a0, a1, bw0, bw1);
      v8f pri = wmma_f32_16x16x4(a0, bw0, vzero);   // fresh priors tile
      pri     = wmma_f32_16x16x4(a1, bw1, pri);
#pragma unroll
      for (int j = 0; j < 8; ++j) {
        const float t = row_sum16(pri[j] * outv[j]);   // DPP16 row reduction
        if (col == 0) {          // lanes 0 and 16 own rows j and j+8
          const int b = j + 8 * grp;
          const float nl = first ? t : (logits[b * R_NODES + r] + t);
          logits[b * R_NODES + r] = nl;
          lmax[j] = fmaxf(lmax[j], nl);   // fused running max
        }
      }
    }
    if (col == 0) {
#pragma unroll
      for (int j = 0; j < 8; ++j) stmp[wave * BT + 8 * grp + j] = lmax[j];
    }
    __syncthreads();
    if (tid < BT) {   // global max per batch row
      float mx = -3.4e38f;
#pragma unroll
      for (int w = 0; w < WAVES; ++w) mx = fmaxf(mx, stmp[w * BT + tid]);
      smax[tid] = mx;
    }
    __syncthreads();
    {   // exp-sum over the R axis
      const int b = tid & 15, seg = tid >> 4;
      const float mx = smax[b];
      float sum = 0.f;
      for (int r = seg * RSEG; r < seg * RSEG + RSEG; ++r)
        sum += __expf(logits[b * R_NODES + r] - mx);
      stmp[seg * BT + b] = sum;
    }
    __syncthreads();
    if (tid < BT) {
      float s = 0.f;
#pragma unroll
      for (int sg = 0; sg < SEGS; ++sg) s += stmp[sg * BT + tid];
      sinv[tid] = 1.0f / s;
    }
    __syncthreads();
  };

  // ---- 3 routing iterations, fully specialized --------------------------
  reduce_squash(sum_pass(true));    // iter 0: uniform probs = 1/R
  delta_pass(true);
  reduce_squash(sum_pass(false));   // iter 1
  delta_pass(false);
  reduce_squash(sum_pass(false));   // iter 2 (final outputs)

  // out[n, b0 + b, o]  with tid = b*16 + o
  out[((size_t)n * BATCH + b0 + (tid >> 4)) * CO + (tid & 15)] = sout[tid];
}

extern "C" void kernel_launch(void* const* d_in, const int* in_sizes, int n_in,
                              void* d_out, int out_size, void* d_ws, size_t ws_size,
                              hipStream_t stream) {
  (void)in_sizes; (void)n_in; (void)out_size; (void)d_ws; (void)ws_size;
  const float* x  = (const float*)d_in[0];   // [256, 1152, 8]
  const float* rw = (const float*)d_in[1];   // [10, 1152, 8, 16]
  float* out = (float*)d_out;                // [10, 256, 1, 1, 16]
  dim3 grid(NCAPS * (BATCH / BT));           // 160 workgroups
  dim3 block(WAVES * 32);                    // 8 wave32 waves
  capsule_route_kernel<<<grid, block, 0, stream>>>(x, rw, out);
}